// MedianBlur_72481868087766
// MI455X (gfx1250) — compile-verified
//
#include <hip/hip_runtime.h>
#include <hip/hip_bf16.h>
#include <stdint.h>

// Median blur 5x5, zero padding, lower-median (index 12 of 25) — matches
// kornia/jnp.sort(...)[12] for odd k. Input [8,3,512,512] f32.

#define H 512
#define W 512
#define TILE_W 32
#define TILE_H 32
#define HALO 2
#define SMW (TILE_W + 2 * HALO) /* 36 */
#define NTHREADS 256

// gfx1250 async global->LDS path (ASYNCcnt). Guarded so host pass and any
// toolchain lacking the builtin still compile via the plain-load fallback.
#if defined(__HIP_DEVICE_COMPILE__) && defined(__has_builtin)
#if __has_builtin(__builtin_amdgcn_global_load_async_to_lds_b32)
#define HAVE_ASYNC_LDS 1
#endif
#endif

typedef __attribute__((address_space(1))) int gint_t;
typedef __attribute__((address_space(3))) int lint_t;

__device__ __forceinline__ void cswap(float& x, float& y) {
  float lo = fminf(x, y);
  float hi = fmaxf(x, y);
  x = lo;
  y = hi;
}

// Forgetful selection median-of-25 with pairwise minmax:
// per step over k live elements, min+max found in ceil(3k/2)-2 compare-
// exchanges (vs 2k-3 for bubble passes). 132 CEs total vs 168.
// t0 points at the top-left tap of the 5x5 window in LDS.
__device__ __forceinline__ float median25(const float* __restrict__ t0,
                                          const int stride) {
  float a[14];
#pragma unroll
  for (int t = 0; t < 14; ++t) a[t] = t0[(t / 5) * stride + (t % 5)];

  int lo = 0;
  const int hi = 13;
#pragma unroll
  for (int t = 14; t < 25; ++t) {
    // --- pairwise minmax over a[lo..hi] ---
    // 1) order adjacent pairs (odd count leaves a[hi] unpaired)
#pragma unroll
    for (int i = lo; i + 1 <= hi; i += 2) cswap(a[i], a[i + 1]);
    // 2) min-chain over low (even-offset) slots -> min at a[lo]
#pragma unroll
    for (int j = lo + 2; j <= hi; j += 2) cswap(a[lo], a[j]);
    // 3) max-chain over high (odd-offset) slots -> max at a[hi]
#pragma unroll
    for (int j = lo + 1; j < hi; j += 2) cswap(a[j], a[hi]);

    ++lo;                                    // forget the min
    a[hi] = t0[(t / 5) * stride + (t % 5)];  // replace the max with next tap
  }
  // three candidates left: a[11], a[12], a[13]; median is the middle one
  cswap(a[11], a[12]);
  cswap(a[12], a[13]);
  cswap(a[11], a[12]);
  return a[12];
}

__global__ __launch_bounds__(NTHREADS) void median5x5_kernel(
    const float* __restrict__ in, float* __restrict__ out) {
  __shared__ float smem[SMW * SMW];

  const int tx = threadIdx.x;            // 0..31
  const int ty = threadIdx.y;            // 0..7
  const int tid = ty * TILE_W + tx;      // 0..255
  const int x0 = blockIdx.x * TILE_W;
  const int y0 = blockIdx.y * TILE_H;
  const size_t plane = (size_t)blockIdx.z * (size_t)(H * W);
  const float* __restrict__ src = in + plane;

  // ---- stage (TILE+2*HALO)^2 input window into LDS, zero-padded ----
  for (int idx = tid; idx < SMW * SMW; idx += NTHREADS) {
    const int r = idx / SMW;
    const int c = idx - r * SMW;
    const int gy = y0 - HALO + r;
    const int gx = x0 - HALO + c;
    float* sp = &smem[idx];
    if (gy >= 0 && gy < H && gx >= 0 && gx < W) {
#if defined(HAVE_ASYNC_LDS)
      __builtin_amdgcn_global_load_async_to_lds_b32(
          (gint_t*)const_cast<float*>(src + (size_t)gy * W + gx),
          (lint_t*)sp,
          /*imm offset*/ 0, /*cpol*/ 0);
#else
      *sp = src[(size_t)gy * W + gx];
#endif
    } else {
      *sp = 0.0f;  // zero padding
    }
  }

#if defined(HAVE_ASYNC_LDS)
#if __has_builtin(__builtin_amdgcn_s_wait_asynccnt)
  __builtin_amdgcn_s_wait_asynccnt(0);
#else
  asm volatile("s_wait_asynccnt 0" ::: "memory");
#endif
#endif
  __syncthreads();

  // ---- compute 4 output pixels per thread ----
  float* __restrict__ dst = out + plane;
  for (int k = 0; k < TILE_H / 8; ++k) {
    const int sy = ty + k * 8;  // output row within tile, 0..31
    const float m = median25(&smem[sy * SMW + tx], SMW);
    dst[(size_t)(y0 + sy) * W + (x0 + tx)] = m;
  }
}

extern "C" void kernel_launch(void* const* d_in, const int* in_sizes, int n_in,
                              void* d_out, int out_size, void* d_ws,
                              size_t ws_size, hipStream_t stream) {
  (void)n_in;
  (void)out_size;
  (void)d_ws;
  (void)ws_size;
  const float* img = (const float*)d_in[0];
  // d_in[1] is kernel_size == 5; baked in at compile time.
  float* out = (float*)d_out;
  const int planes = in_sizes[0] / (H * W);  // B*C = 24
  dim3 block(TILE_W, 8, 1);
  dim3 grid(W / TILE_W, H / TILE_H, planes);
  median5x5_kernel<<<grid, block, 0, stream>>>(img, out);
}